// LIFSpike_vanilla_84954453115681
// MI455X (gfx1250) — compile-verified
//
#include <hip/hip_runtime.h>

// LIF spike recurrence over T=8 timesteps.
//   u_t = TAU * u_{t-1} * (1 - o_{t-1}) + x_t
//   o_t = (u_t - VTH > 0) ? 1 : 0
// x, out: [T, B, C, H, W] f32, T*N elements with N = B*C*H*W.
// Memory-bound: 512 MB total traffic, ~22 us floor at 23.3 TB/s.

typedef float v4f __attribute__((ext_vector_type(4)));

#define LIF_TAU 0.5f
#define LIF_VTH 0.99999f
#define LIF_T   8

__global__ __launch_bounds__(256) void lif_spike_kernel(
    const float* __restrict__ x_,
    float* __restrict__ out_,
    long long n4)   // number of float4 chunks per timestep (N/4)
{
    long long i = (long long)blockIdx.x * blockDim.x + threadIdx.x;
    if (i >= n4) return;

    const v4f* __restrict__ x   = (const v4f*)x_;
    v4f*       __restrict__ out = (v4f*)out_;

    // All T loads are independent of the recurrence: issue them all up front
    // (8 outstanding global_load_b128 per thread for deep MLP). Non-temporal
    // hint: single-touch stream, keep it out of WGP$/L2 (CDNA5 TH=NT).
    v4f xs[LIF_T];
#pragma unroll
    for (int t = 0; t < LIF_T; ++t) {
        xs[t] = __builtin_nontemporal_load(&x[(long long)t * n4 + i]);
    }

    v4f u = (v4f)0.0f;   // membrane potential
    v4f o = (v4f)0.0f;   // previous spike

#pragma unroll
    for (int t = 0; t < LIF_T; ++t) {
        // u_new = TAU * u * (1 - o_prev) + x_t   (vector FMA-friendly form)
        u = LIF_TAU * u * (1.0f - o) + xs[t];

        // heaviside(u - VTH): exact match to reference's (u_new - VTH > 0)
        v4f d = u - LIF_VTH;
        v4f s;
        s.x = (d.x > 0.0f) ? 1.0f : 0.0f;
        s.y = (d.y > 0.0f) ? 1.0f : 0.0f;
        s.z = (d.z > 0.0f) ? 1.0f : 0.0f;
        s.w = (d.w > 0.0f) ? 1.0f : 0.0f;
        o = s;

        // Stream the spike out (non-temporal b128 store) as soon as ready.
        __builtin_nontemporal_store(o, &out[(long long)t * n4 + i]);
    }
}

extern "C" void kernel_launch(void* const* d_in, const int* in_sizes, int n_in,
                              void* d_out, int out_size, void* d_ws, size_t ws_size,
                              hipStream_t stream) {
    (void)n_in; (void)d_ws; (void)ws_size; (void)out_size;

    const float* x = (const float*)d_in[0];
    float* out     = (float*)d_out;

    // in_sizes[0] = T * B * C * H * W total elements; T is fixed at 8 by the
    // reference. Per-timestep element count N, vectorized by 4.
    long long total = (long long)in_sizes[0];
    long long n     = total / LIF_T;      // 8,388,608
    long long n4    = n / 4;              // 2,097,152 float4 chunks (N % 4 == 0)

    const int block = 256;                // 8 wave32s per workgroup
    long long grid  = (n4 + block - 1) / block;  // 8192 blocks

    lif_spike_kernel<<<(dim3)(unsigned)grid, block, 0, stream>>>(x, out, n4);
}